// AdderConv_75265006895564
// MI455X (gfx1250) — compile-verified
//
#include <hip/hip_runtime.h>
#include <stdint.h>

#define N_   8
#define C_   32
#define H_   28
#define WD_  28
#define O_   64
#define KDIM 288              // C_*3*3
#define HPW  36               // padded LDS row stride (144 B = 9*16 B)
#define XOFF 4                // interior (input col 0) starts at float offset 4 (16B aligned)
                              // halo zeros live at offsets 3 and 32
#define BAND 7                // output rows per workgroup
#define LR   9                // input rows staged per workgroup (BAND+2)
#define OTC  16               // output channels per workgroup
#define XS_ELEMS (C_*LR*HPW)  // 10368 floats = 41472 B
#define WS_ELEMS (OTC*KDIM)   // 4608 floats  = 18432 B
#define PLANE (H_*WD_)        // 784
#define CNT_PER_CH (N_*PLANE) // 6272
#define EPSV 1e-5f

// ---- CDNA5 async global->LDS copies (ASYNCcnt-tracked) ------------------
__device__ __forceinline__ void async_ld_b128(uint32_t lds_off, const void* g) {
    uint64_t ga = (uint64_t)(uintptr_t)g;
    asm volatile("global_load_async_to_lds_b128 %0, %1, off"
                 :: "v"(lds_off), "v"(ga) : "memory");
}
__device__ __forceinline__ void wait_async0() {
    asm volatile("s_wait_asynccnt 0" ::: "memory");
}

// ---- Kernel 1: AdderNet conv, raw output to d_out -----------------------
__global__ __launch_bounds__(256)
void adder_conv_kernel(const float* __restrict__ x,
                       const float* __restrict__ Wt,
                       float* __restrict__ out)
{
    __shared__ __align__(16) float x_s[XS_ELEMS];
    __shared__ __align__(16) float w_s[WS_ELEMS];

    const int tid = threadIdx.x;
    const int b   = blockIdx.x;        // n*16 + ot*4 + rb
    const int n   = b >> 4;
    const int ot  = (b >> 2) & 3;
    const int rb  = b & 3;
    const int r0  = rb * BAND;         // first output row of this band
    const int o0  = ot * OTC;          // first output channel of this tile

    // Zero the LDS x-slab (covers halo padding + out-of-image rows).
    // Must fully land before any async LDS writes are issued (DS-write vs
    // async-write ordering is not architecturally guaranteed), so there is a
    // barrier between the zeroing and the async issue.
    for (int i = tid; i < XS_ELEMS; i += 256) x_s[i] = 0.0f;
    __syncthreads();

    const uint32_t xs_base = (uint32_t)(uintptr_t)(&x_s[0]);
    const uint32_t ws_base = (uint32_t)(uintptr_t)(&w_s[0]);

    // Weights tile: 16*288 floats, fully contiguous in global -> b128 copies.
    {
        const float* wsrc = Wt + (size_t)o0 * KDIM;
        for (int i = tid; i < WS_ELEMS / 4; i += 256)
            async_ld_b128(ws_base + (uint32_t)i * 16u, wsrc + i * 4);
    }
    // Input slab: 32 ch x up-to-9 rows x 28 cols. Each global row is
    // 112 B = 7 x 16 B and lands at LDS float offset XOFF (16B aligned),
    // so the whole slab moves as b128 async copies: 32*9*7 = 2016 items.
    {
        const float* xsrc = x + (size_t)n * C_ * H_ * WD_;
        for (int i = tid; i < C_ * LR * (WD_ / 4); i += 256) {
            int c   = i / (LR * (WD_ / 4));
            int rm  = i % (LR * (WD_ / 4));
            int lr  = rm / (WD_ / 4);
            int seg = rm % (WD_ / 4);          // 16B segment within the row
            int gr  = r0 - 1 + lr;
            if (gr >= 0 && gr < H_) {
                uint32_t doff = (uint32_t)(((c * LR + lr) * HPW + XOFF + seg * 4) * 4);
                async_ld_b128(xs_base + doff, xsrc + (c * H_ + gr) * WD_ + seg * 4);
            }
        }
    }
    wait_async0();
    __syncthreads();

    // Compute: 4 o-groups x (7 rows x 28 cols) = 784 work items.
    // Each item: one spatial position, 4 consecutive output channels
    // (1 LDS x-load amortized over 4 accumulators).
    for (int e = tid; e < 4 * BAND * WD_; e += 256) {
        const int og  = e / (BAND * WD_);
        const int s   = e % (BAND * WD_);
        const int row = s / WD_;
        const int col = s % WD_;
        const int ob  = og * 4;

        float a0 = 0.f, a1 = 0.f, a2 = 0.f, a3 = 0.f;
        for (int c = 0; c < C_; ++c) {
#pragma unroll
            for (int kh = 0; kh < 3; ++kh) {
                // input col (col-1+kw) lives at padded offset XOFF-1+col+kw
                const float* xr  = &x_s[(c * LR + row + kh) * HPW + (XOFF - 1) + col];
                const float* wr0 = &w_s[(ob + 0) * KDIM + c * 9 + kh * 3];
                const float* wr1 = wr0 + KDIM;
                const float* wr2 = wr1 + KDIM;
                const float* wr3 = wr2 + KDIM;
#pragma unroll
                for (int kw = 0; kw < 3; ++kw) {
                    const float xv = xr[kw];
                    a0 += __builtin_fabsf(xv - wr0[kw]);
                    a1 += __builtin_fabsf(xv - wr1[kw]);
                    a2 += __builtin_fabsf(xv - wr2[kw]);
                    a3 += __builtin_fabsf(xv - wr3[kw]);
                }
            }
        }
        const int gr = r0 + row;
        size_t base = ((size_t)n * O_ + (o0 + ob)) * PLANE + (size_t)gr * WD_ + col;
        out[base]             = -a0;
        out[base + 1 * PLANE] = -a1;
        out[base + 2 * PLANE] = -a2;
        out[base + 3 * PLANE] = -a3;
    }
}

// ---- Kernel 2: per-channel BN stats (deterministic tree reduction) ------
// ws layout (floats): [128,192) scale, [192,256) bias
__global__ __launch_bounds__(256)
void bn_stats_kernel(const float* __restrict__ out,
                     const float* __restrict__ gamma,
                     const float* __restrict__ beta,
                     float* __restrict__ ws)
{
    __shared__ float rs[256];
    __shared__ float rq[256];
    const int o   = blockIdx.x;   // one block per channel
    const int tid = threadIdx.x;

    float s = 0.f, q = 0.f;
    for (int i = tid; i < CNT_PER_CH; i += 256) {
        const int n  = i / PLANE;
        const int sp = i % PLANE;
        const float v = out[((size_t)n * O_ + o) * PLANE + sp];
        s += v;
        q += v * v;
    }
    rs[tid] = s; rq[tid] = q;
    __syncthreads();
    for (int off = 128; off > 0; off >>= 1) {
        if (tid < off) { rs[tid] += rs[tid + off]; rq[tid] += rq[tid + off]; }
        __syncthreads();
    }
    if (tid == 0) {
        const float inv  = 1.0f / (float)CNT_PER_CH;
        const float mean = rs[0] * inv;
        const float var  = rq[0] * inv - mean * mean;   // biased variance
        const float sc   = gamma[o] * rsqrtf(var + EPSV);
        ws[128 + o] = sc;
        ws[192 + o] = beta[o] - mean * sc;
    }
}

// ---- Kernel 3: apply BN in place on d_out -------------------------------
__global__ __launch_bounds__(256)
void bn_apply_kernel(float* __restrict__ out, const float* __restrict__ ws)
{
    const int idx = blockIdx.x * 256 + threadIdx.x;
    const int total = N_ * O_ * PLANE;
    if (idx < total) {
        const int o = (idx / PLANE) % O_;
        out[idx] = out[idx] * ws[128 + o] + ws[192 + o];
    }
}

extern "C" void kernel_launch(void* const* d_in, const int* in_sizes, int n_in,
                              void* d_out, int out_size, void* d_ws, size_t ws_size,
                              hipStream_t stream) {
    (void)in_sizes; (void)n_in; (void)out_size; (void)ws_size;
    const float* x     = (const float*)d_in[0];
    const float* Wt    = (const float*)d_in[1];
    const float* gamma = (const float*)d_in[2];
    const float* beta  = (const float*)d_in[3];
    float* out = (float*)d_out;
    float* ws  = (float*)d_ws;

    // 8 images * 4 channel-tiles * 4 row-bands = 128 workgroups
    adder_conv_kernel<<<dim3(N_ * 4 * 4), dim3(256), 0, stream>>>(x, Wt, out);
    bn_stats_kernel<<<dim3(O_), dim3(256), 0, stream>>>(out, gamma, beta, ws);
    const int total = N_ * O_ * PLANE;
    bn_apply_kernel<<<dim3((total + 255) / 256), dim3(256), 0, stream>>>(out, ws);
}